// GAT_45011257262604
// MI455X (gfx1250) — compile-verified
//
#include <hip/hip_runtime.h>
#include <hip/hip_bf16.h>

typedef float v2f __attribute__((ext_vector_type(2)));
typedef float v8f __attribute__((ext_vector_type(8)));

#define HC 256
#define SLOPE 0.2f

static inline int cdiv(int a, int b) { return (a + b - 1) / b; }

// ---------------- utility fills ----------------
__global__ void k_fill(float* __restrict__ p, float v, int n) {
    int i = blockIdx.x * blockDim.x + threadIdx.x;
    if (i < n) p[i] = v;
}

// ---------------- self-loop attr: cnt + weighted sum ----------------
__global__ void k_count(const int* __restrict__ dst, const float* __restrict__ ew,
                        float* __restrict__ cnt, float* __restrict__ lsum, int E) {
    int e = blockIdx.x * blockDim.x + threadIdx.x;
    if (e >= E) return;
    int d = dst[e];
    atomicAdd(&cnt[d], 1.0f);
    atomicAdd(&lsum[d], ew[e]);
}

__global__ void k_loopattr(float* __restrict__ lsum, const float* __restrict__ cnt, int N) {
    int n = blockIdx.x * blockDim.x + threadIdx.x;
    if (n >= N) return;
    lsum[n] = lsum[n] / fmaxf(cnt[n], 1.0f);
}

// ---------------- edge-attr attention coefficient: coef[h] = sum_c We[h*64+c]*ae[h*64+c]
__global__ void k_coef(const float* __restrict__ We, const float* __restrict__ ae,
                       float* __restrict__ coef) {
    __shared__ float red[HC];
    int t = threadIdx.x;
    red[t] = We[t] * ae[t];
    __syncthreads();
    if (t < 4) {
        float s = 0.f;
        #pragma unroll
        for (int c = 0; c < 64; ++c) s += red[t * 64 + c];
        coef[t] = s;
    }
}

// ---------------- WMMA f32 GEMM: Hout[M x 256] = X[M x K] @ W[K x 256] ----------------
// One wave32 computes one 16x16 tile via V_WMMA_F32_16X16X4_F32, stepping K by 4.
__global__ void __launch_bounds__(256)
k_gemm_wmma(const float* __restrict__ X, const float* __restrict__ W,
            float* __restrict__ Hout, int M, int K) {
    const int wave = threadIdx.x >> 5;
    const int lane = threadIdx.x & 31;
    const int wid  = blockIdx.x * 8 + wave;
    const int ntile = wid & 15;           // 256/16 = 16 tiles along N
    const int mtile = wid >> 4;
    const int m0 = mtile * 16;
    if (m0 >= M) return;
    const int n0 = ntile * 16;
    const int lhalf = lane >> 4;          // 0: K pair {0,1}; 1: K pair {2,3}
    const int l = lane & 15;

    // A 16x4 f32 layout: lane<16 -> row l, K = {0,1}; lane>=16 -> row l, K = {2,3}
    const float* Arow = X + (size_t)(m0 + l) * K + 2 * lhalf;
    // B 4x16 f32 (mirrored layout): lane<16 -> col l, K rows {0,1}; lane>=16 -> K rows {2,3}
    const float* Bcol = W + (size_t)(2 * lhalf) * HC + (n0 + l);

    v8f acc = {0.f, 0.f, 0.f, 0.f, 0.f, 0.f, 0.f, 0.f};
    for (int k = 0; k < K; k += 4) {
        v2f a, b;
        a.x = Arow[k + 0];
        a.y = Arow[k + 1];
        b.x = Bcol[(size_t)k * HC];
        b.y = Bcol[(size_t)k * HC + HC];
        acc = __builtin_amdgcn_wmma_f32_16x16x4_f32(
            /*neg_a=*/false, a, /*neg_b=*/false, b,
            /*c_mod=*/(short)0, acc, /*reuse_a=*/false, /*reuse_b=*/false);
    }
    // C/D layout: VGPR r -> row m0 + r (lanes 0-15) or m0 + 8 + r (lanes 16-31), col n0 + l
    float* out = Hout + (size_t)(m0 + 8 * lhalf) * HC + n0 + l;
    #pragma unroll
    for (int r = 0; r < 8; ++r) out[(size_t)r * HC] = acc[r];
}

// ---------------- per-node attention scores ----------------
__global__ void k_alpha_nodes(const float* __restrict__ h, const float* __restrict__ a_s,
                              const float* __restrict__ a_d, float* __restrict__ as_out,
                              float* __restrict__ ad_out, int N) {
    int t = blockIdx.x * blockDim.x + threadIdx.x;
    if (t >= N * 4) return;
    int head = t & 3, n = t >> 2;
    const float* hp = h + (size_t)n * HC + head * 64;
    const float* sp = a_s + head * 64;
    const float* dp = a_d + head * 64;
    float ss = 0.f, sd = 0.f;
    #pragma unroll 8
    for (int c = 0; c < 64; ++c) { float v = hp[c]; ss += v * sp[c]; sd += v * dp[c]; }
    as_out[t] = ss;
    ad_out[t] = sd;
}

// ---------------- CAS-based float atomic max ----------------
__device__ inline void atomicMaxF(float* addr, float val) {
    unsigned int* ua = (unsigned int*)addr;
    unsigned int old = __float_as_uint(-INFINITY);
    old = *ua;
    while (__uint_as_float(old) < val) {
        unsigned int assumed = old;
        old = atomicCAS(ua, assumed, __float_as_uint(val));
        if (old == assumed) break;
    }
}

// ---------------- per-edge alpha + segment max ----------------
__global__ void k_alpha_edges(const int* __restrict__ src, const int* __restrict__ dst,
                              const float* __restrict__ ew, const float* __restrict__ loop_attr,
                              const float* __restrict__ as_, const float* __restrict__ ad_,
                              const float* __restrict__ coef, float* __restrict__ alpha,
                              float* __restrict__ m, int E, int Etot) {
    int e = blockIdx.x * blockDim.x + threadIdx.x;
    if (e >= Etot) return;
    int s, d; float w;
    if (e < E) { s = src[e]; d = dst[e]; w = ew[e]; }
    else       { s = d = e - E; w = loop_attr[s]; }
    #pragma unroll
    for (int hd = 0; hd < 4; ++hd) {
        float a = as_[s * 4 + hd] + ad_[d * 4 + hd] + w * coef[hd];
        a = (a > 0.f) ? a : SLOPE * a;      // leaky relu
        alpha[(size_t)e * 4 + hd] = a;
        atomicMaxF(&m[d * 4 + hd], a);
    }
}

__global__ void k_fix_m(float* __restrict__ m, int n) {
    int i = blockIdx.x * blockDim.x + threadIdx.x;
    if (i >= n) return;
    float v = m[i];
    if (!isfinite(v)) m[i] = 0.f;
}

// ---------------- per-edge exp + segment sum ----------------
__global__ void k_exp_edges(const int* __restrict__ dst, float* __restrict__ alpha_ex,
                            const float* __restrict__ m, float* __restrict__ ssum,
                            int E, int Etot) {
    int e = blockIdx.x * blockDim.x + threadIdx.x;
    if (e >= Etot) return;
    int d = (e < E) ? dst[e] : (e - E);
    #pragma unroll
    for (int hd = 0; hd < 4; ++hd) {
        float ex = expf(alpha_ex[(size_t)e * 4 + hd] - m[d * 4 + hd]);
        alpha_ex[(size_t)e * 4 + hd] = ex;
        atomicAdd(&ssum[d * 4 + hd], ex);
    }
}

// ---------------- scatter aggregation: out[d,:] += ex[e,head] * h[s,:] ----------------
__global__ void __launch_bounds__(256)
k_aggregate(const int* __restrict__ src, const int* __restrict__ dst,
            const float* __restrict__ ex, const float* __restrict__ h,
            float* __restrict__ out, int E) {
    int e = blockIdx.x;
    int c = threadIdx.x;
    int s, d;
    if (e < E) { s = src[e]; d = dst[e]; }
    else       { s = d = e - E; }
    float w = ex[(size_t)e * 4 + (c >> 6)];
    atomicAdd(&out[(size_t)d * HC + c], w * h[(size_t)s * HC + c]);
}

// ---------------- normalize + bias ----------------
__global__ void k_finalize(float* __restrict__ out, const float* __restrict__ ssum,
                           const float* __restrict__ b, int N) {
    int t = blockIdx.x * blockDim.x + threadIdx.x;
    if (t >= N * HC) return;
    int n = t >> 8, c = t & 255;
    out[t] = out[t] / (ssum[n * 4 + (c >> 6)] + 1e-16f) + b[c];
}

extern "C" void kernel_launch(void* const* d_in, const int* in_sizes, int n_in,
                              void* d_out, int out_size, void* d_ws, size_t ws_size,
                              hipStream_t stream) {
    const float* x   = (const float*)d_in[0];
    const int*   ei  = (const int*)d_in[1];
    const float* ew  = (const float*)d_in[2];
    const float* W1  = (const float*)d_in[3];
    const float* as1 = (const float*)d_in[4];
    const float* ad1 = (const float*)d_in[5];
    const float* We1 = (const float*)d_in[6];
    const float* ae1 = (const float*)d_in[7];
    const float* b1  = (const float*)d_in[8];
    const float* W2  = (const float*)d_in[9];
    const float* as2 = (const float*)d_in[10];
    const float* ad2 = (const float*)d_in[11];
    const float* We2 = (const float*)d_in[12];
    const float* ae2 = (const float*)d_in[13];
    const float* b2  = (const float*)d_in[14];

    const int IN_DIM = 512;
    const int N = in_sizes[0] / IN_DIM;
    const int E = in_sizes[2];
    const int Etot = E + N;
    const int* src = ei;
    const int* dst = ei + E;

    // workspace layout (floats)
    float* f        = (float*)d_ws;
    float* cnt      = f;  f += N;
    float* loopat   = f;  f += N;          // loop_sum then loop_attr (in place)
    float* h        = f;  f += (size_t)N * HC;
    float* alpha_s  = f;  f += (size_t)N * 4;
    float* alpha_d  = f;  f += (size_t)N * 4;
    float* m        = f;  f += (size_t)N * 4;
    float* ssum     = f;  f += (size_t)N * 4;
    float* exbuf    = f;  f += (size_t)Etot * 4;
    float* coef     = f;  f += 4;

    float* out1 = (float*)d_out;   // layer-1 output lives in d_out, consumed by layer-2 GEMM
    float* out2 = (float*)d_out;

    const int B = 256;

    // --- self-loop edge attribute ---
    k_fill<<<cdiv(N, B), B, 0, stream>>>(cnt, 0.f, N);
    k_fill<<<cdiv(N, B), B, 0, stream>>>(loopat, 0.f, N);
    k_count<<<cdiv(E, B), B, 0, stream>>>(dst, ew, cnt, loopat, E);
    k_loopattr<<<cdiv(N, B), B, 0, stream>>>(loopat, cnt, N);

    const int gemm_blocks = cdiv((N / 16) * 16, 8);   // waves = Mtiles*16 Ntiles, 8 waves/block

    // ================= layer 1 =================
    k_coef<<<1, 256, 0, stream>>>(We1, ae1, coef);
    k_gemm_wmma<<<gemm_blocks, 256, 0, stream>>>(x, W1, h, N, 512);
    k_alpha_nodes<<<cdiv(N * 4, B), B, 0, stream>>>(h, as1, ad1, alpha_s, alpha_d, N);
    k_fill<<<cdiv(N * 4, B), B, 0, stream>>>(m, -INFINITY, N * 4);
    k_fill<<<cdiv(N * 4, B), B, 0, stream>>>(ssum, 0.f, N * 4);
    k_alpha_edges<<<cdiv(Etot, B), B, 0, stream>>>(src, dst, ew, loopat, alpha_s, alpha_d,
                                                   coef, exbuf, m, E, Etot);
    k_fix_m<<<cdiv(N * 4, B), B, 0, stream>>>(m, N * 4);
    k_exp_edges<<<cdiv(Etot, B), B, 0, stream>>>(dst, exbuf, m, ssum, E, Etot);
    k_fill<<<cdiv(N * HC, B), B, 0, stream>>>(out1, 0.f, N * HC);
    k_aggregate<<<Etot, 256, 0, stream>>>(src, dst, exbuf, h, out1, E);
    k_finalize<<<cdiv(N * HC, B), B, 0, stream>>>(out1, ssum, b1, N);

    // ================= layer 2 =================
    k_coef<<<1, 256, 0, stream>>>(We2, ae2, coef);
    k_gemm_wmma<<<gemm_blocks, 256, 0, stream>>>(out1, W2, h, N, 256);  // consumes d_out
    k_alpha_nodes<<<cdiv(N * 4, B), B, 0, stream>>>(h, as2, ad2, alpha_s, alpha_d, N);
    k_fill<<<cdiv(N * 4, B), B, 0, stream>>>(m, -INFINITY, N * 4);
    k_fill<<<cdiv(N * 4, B), B, 0, stream>>>(ssum, 0.f, N * 4);
    k_alpha_edges<<<cdiv(Etot, B), B, 0, stream>>>(src, dst, ew, loopat, alpha_s, alpha_d,
                                                   coef, exbuf, m, E, Etot);
    k_fix_m<<<cdiv(N * 4, B), B, 0, stream>>>(m, N * 4);
    k_exp_edges<<<cdiv(Etot, B), B, 0, stream>>>(dst, exbuf, m, ssum, E, Etot);
    k_fill<<<cdiv(N * HC, B), B, 0, stream>>>(out2, 0.f, N * HC);      // safe: GEMM2 already read it
    k_aggregate<<<Etot, 256, 0, stream>>>(src, dst, exbuf, h, out2, E);
    k_finalize<<<cdiv(N * HC, B), B, 0, stream>>>(out2, ssum, b2, N);
}